// ISDALoss_4767413698904
// MI455X (gfx1250) — compile-verified
//
#include <hip/hip_runtime.h>
#include <hip/hip_bf16.h>

typedef __attribute__((ext_vector_type(2))) float v2f;
typedef __attribute__((ext_vector_type(8))) float v8f;

#define KN   4
#define KA   256
#define KH   256
#define KW   256
#define KC   19
#define KCN  20
#define KHW  (KH * KW)          // 65536
#define ASTR 257                // padded LDS row stride (bank-conflict-free)

// ---------------------------------------------------------------- kernel 0
__global__ void zero_ws_kernel(float* ws, int n) {
    for (int i = blockIdx.x * 256 + threadIdx.x; i < n; i += gridDim.x * 256)
        ws[i] = 0.f;
}

// ---------------------------------------------------------------- kernel 1
// counts[k] = #pixels with downsampled label k (k<19; ignore 255)
__global__ void count_kernel(const int* __restrict__ tx, float* __restrict__ cnt) {
    __shared__ float lc[KCN];
    int tid = threadIdx.x;
    if (tid < KCN) lc[tid] = 0.f;
    __syncthreads();
    int base = blockIdx.x * 1024;
    #pragma unroll
    for (int i = 0; i < 4; ++i) {
        int idx = base + i * 256 + tid;      // pixel in [0, N*HW)
        int n = idx >> 16;
        int p = idx & (KHW - 1);
        int h = p >> 8, w = p & 255;
        int t = tx[n * (4 * KHW) + (h << 10) + (w << 1)];   // target_x[n,2h,2w]
        if (t != 255) atomicAdd(&lc[t], 1.f);
    }
    __syncthreads();
    if (tid < KCN) atomicAdd(&cnt[tid], lc[tid]);
}

// ---------------------------------------------------------------- kernel 2
// per-(class,channel) sum and sum-of-squares of features (segment statistics)
// grid.x = n*A plane id, grid.y = quarter of the HW plane
// Each thread processes 4 consecutive pixels -> global_load_b128.
__global__ void accum_kernel(const float* __restrict__ feat,
                             const int* __restrict__ tx,
                             float* __restrict__ gsum, float* __restrict__ gsumsq) {
    __shared__ float ls[KCN];
    __shared__ float lq[KCN];
    int tid = threadIdx.x;
    if (tid < KCN) { ls[tid] = 0.f; lq[tid] = 0.f; }
    __syncthreads();

    int na = blockIdx.x;                    // n*256 + a
    int n  = na >> 8;
    const float* plane = feat + (size_t)na * KHW;
    const int*   txn   = tx + n * (4 * KHW);
    int base = blockIdx.y * (KHW / 4);      // 16384 pixels per block

    #pragma unroll 2
    for (int i = 0; i < 16; ++i) {          // 16 iters x 256 thr x 4 px = 16384
        int idx = base + (i * 256 + tid) * 4;           // multiple of 4
        __builtin_prefetch(plane + idx + 16384, 0, 0);  // global_prefetch_b8
        int h = idx >> 8, w = idx & 255;                // w multiple of 4
        int to = (h << 10) + (w << 1);                  // multiple of 8 ints
        float4 f  = *(const float4*)(plane + idx);
        int4   ta = *(const int4*)(txn + to);
        int4   tb = *(const int4*)(txn + to + 4);
        int t0 = ta.x, t1 = ta.z, t2 = tb.x, t3 = tb.z; // target_x[n,2h,2(w+j)]
        if (t0 != 255) { atomicAdd(&ls[t0], f.x); atomicAdd(&lq[t0], f.x * f.x); }
        if (t1 != 255) { atomicAdd(&ls[t1], f.y); atomicAdd(&lq[t1], f.y * f.y); }
        if (t2 != 255) { atomicAdd(&ls[t2], f.z); atomicAdd(&lq[t2], f.z * f.z); }
        if (t3 != 255) { atomicAdd(&ls[t3], f.w); atomicAdd(&lq[t3], f.w * f.w); }
    }
    __syncthreads();
    int a = na & 255;
    if (tid < KCN) {
        atomicAdd(&gsum[tid * KA + a],   ls[tid]);
        atomicAdd(&gsumsq[tid * KA + a], lq[tid]);
    }
}

// ---------------------------------------------------------------- kernel 3
// Compute CoV (EstimatorCV.update_CV) then the 19x19 sigma2 table S with
// f32 WMMA:  T1 = CoV x (W.*W)^T ,  T2 = (W.*CoV) x W^T ,  T3 = diag term.
__global__ void build_S_kernel(const float* __restrict__ fcw,
                               const float* __restrict__ Ave,
                               const float* __restrict__ CoVarIn,
                               const float* __restrict__ Amount,
                               const float* __restrict__ ratio,
                               const float* __restrict__ gsum,
                               const float* __restrict__ gsumsq,
                               const float* __restrict__ gcnt,
                               float* __restrict__ Sout) {
    __shared__ float sCoV[KCN * ASTR];      // 20 x 256 (padded)
    __shared__ float sW[KC * ASTR];         // 19 x 256 (padded)
    __shared__ float sT1[32 * 32];
    __shared__ float sT2[32 * 32];
    __shared__ float sT3[32];

    int tid = threadIdx.x;

    // fc weights into LDS
    for (int i = tid; i < KC * KA; i += 256) {
        int k = i >> 8, a = i & 255;
        sW[k * ASTR + a] = fcw[i];
    }
    // CoV[k,a] = CoVariance*(1-w) + var*w + w*(1-w)*(Ave-ave)^2
    for (int i = tid; i < KCN * KA; i += 256) {
        int k = i >> 8, a = i & 255;
        float cN    = gcnt[k];
        float denom = fmaxf(cN, 1.f);
        float s   = gsum[i];
        float q   = gsumsq[i];
        float ave = s / denom;
        float var = (q - 2.f * ave * s + cN * ave * ave) / denom;  // segsum((f-ave)^2)/denom
        float amt = Amount[k];
        float wk  = (cN + amt > 0.f) ? (cN / (cN + amt)) : 0.f;    // NaN(0/0) -> 0
        float d   = Ave[i] - ave;
        sCoV[k * ASTR + a] = CoVarIn[i] * (1.f - wk) + var * wk + wk * (1.f - wk) * d * d;
    }
    __syncthreads();

    // 8 waves: waves 0-3 -> four 16x16 tiles of T1 (32x32 padded output),
    // waves 4-7 -> T2. K = 256 via 64 x V_WMMA_F32_16X16X4_F32 per wave.
    int wv   = tid >> 5;
    int lane = tid & 31;
    int job  = wv & 3;
    int mt   = job >> 1;                    // M tile (rows = k)
    int nt   = job & 1;                     // N tile (cols = c)
    bool doT2 = (wv >= 4);                  // wave-uniform

    int mrow = mt * 16 + (lane & 15);       // 0..31 (matrix row k)
    int ncol = nt * 16 + (lane & 15);       // 0..31 (matrix col c)
    int khi  = (lane >> 4) * 2;             // lanes 0-15 -> K 0,1 ; 16-31 -> K 2,3

    int   rA  = (mrow < KCN) ? mrow : 0;    float mA  = (mrow < KCN) ? 1.f : 0.f;
    int   rAw = (mrow < KC)  ? mrow : 0;    float mAw = (mrow < KC)  ? 1.f : 0.f;
    int   rB  = (ncol < KC)  ? ncol : 0;    float mB  = (ncol < KC)  ? 1.f : 0.f;

    v8f acc = {};
    if (doT2) {
        // T2 = (W .* CoV) x W^T   (rows k, cols c)
        float mAcw = mA * mAw;
        for (int kk = 0; kk < KA; kk += 4) {
            int i0 = kk + khi, i1 = i0 + 1;
            float a0 = sCoV[rA * ASTR + i0] * sW[rAw * ASTR + i0] * mAcw;
            float a1 = sCoV[rA * ASTR + i1] * sW[rAw * ASTR + i1] * mAcw;
            float b0 = sW[rB * ASTR + i0] * mB;
            float b1 = sW[rB * ASTR + i1] * mB;
            v2f av = { a0, a1 };
            v2f bv = { b0, b1 };
            acc = __builtin_amdgcn_wmma_f32_16x16x4_f32(
                false, av, false, bv, (short)0, acc, false, false);
        }
    } else {
        // T1 = CoV x (W .* W)^T
        for (int kk = 0; kk < KA; kk += 4) {
            int i0 = kk + khi, i1 = i0 + 1;
            float a0 = sCoV[rA * ASTR + i0] * mA;
            float a1 = sCoV[rA * ASTR + i1] * mA;
            float w0 = sW[rB * ASTR + i0];
            float w1 = sW[rB * ASTR + i1];
            float b0 = w0 * w0 * mB;
            float b1 = w1 * w1 * mB;
            v2f av = { a0, a1 };
            v2f bv = { b0, b1 };
            acc = __builtin_amdgcn_wmma_f32_16x16x4_f32(
                false, av, false, bv, (short)0, acc, false, false);
        }
    }

    // D tile -> LDS (16x16 f32 C/D layout: VGPR r = row r (lanes 0-15) / r+8)
    float* dtile = doT2 ? sT2 : sT1;
    #pragma unroll
    for (int r = 0; r < 8; ++r) {
        int row = mt * 16 + ((lane < 16) ? r : r + 8);
        int col = nt * 16 + (lane & 15);
        dtile[row * 32 + col] = acc[r];
    }
    __syncthreads();

    // T3[k] = sum_a W[k,a]^2 * CoV[k,a]
    if (tid < KC) {
        float t3 = 0.f;
        for (int a = 0; a < KA; ++a) {
            float w = sW[tid * ASTR + a];
            t3 = fmaf(w * w, sCoV[tid * ASTR + a], t3);
        }
        sT3[tid] = t3;
    }
    __syncthreads();

    float rr = ratio[0];
    for (int i = tid; i < KC * KC; i += 256) {
        int k = i / KC, c = i % KC;
        Sout[i] = rr * (sT1[k * 32 + c] - 2.f * sT2[k * 32 + c] + sT3[k]);
    }
}

// ---------------------------------------------------------------- kernel 4
// out[n,c,h,w] = y[n,c,h,w] + 0.5 * S[lab(n,h,w), c] * (lab != 255)
// float4 / int4 vectorized: b128 loads + b128 stores.
__global__ void apply_kernel(const float* __restrict__ y,
                             const int* __restrict__ tx,
                             const float* __restrict__ S,
                             float* __restrict__ out) {
    __shared__ float sS[KC * KC];
    int tid = threadIdx.x;
    for (int i = tid; i < KC * KC; i += 256) sS[i] = S[i];
    __syncthreads();

    int plane = blockIdx.y;                 // n*C + c, 0..75
    int n = plane / KC, c = plane % KC;
    const int* txn = tx + n * (4 * KHW);
    const float* yp = y + (size_t)plane * KHW;
    float* op = out + (size_t)plane * KHW;
    int base = blockIdx.x * 4096;

    #pragma unroll
    for (int i = 0; i < 4; ++i) {           // 4 iters x 256 thr x 4 px = 4096
        int idx = base + (i * 256 + tid) * 4;
        int h = idx >> 8, w = idx & 255;
        int to = (h << 10) + (w << 1);
        float4 yv = *(const float4*)(yp + idx);
        int4   ta = *(const int4*)(txn + to);
        int4   tb = *(const int4*)(txn + to + 4);
        float4 o;
        o.x = yv.x + ((ta.x != 255) ? 0.5f * sS[ta.x * KC + c] : 0.f);
        o.y = yv.y + ((ta.z != 255) ? 0.5f * sS[ta.z * KC + c] : 0.f);
        o.z = yv.z + ((tb.x != 255) ? 0.5f * sS[tb.x * KC + c] : 0.f);
        o.w = yv.w + ((tb.z != 255) ? 0.5f * sS[tb.z * KC + c] : 0.f);
        *(float4*)(op + idx) = o;
    }
}

// ---------------------------------------------------------------- launch
extern "C" void kernel_launch(void* const* d_in, const int* in_sizes, int n_in,
                              void* d_out, int out_size, void* d_ws, size_t ws_size,
                              hipStream_t stream) {
    const float* features  = (const float*)d_in[0];  // [4,256,256,256]
    const float* fc_weight = (const float*)d_in[1];  // [19,256]
    const float* y         = (const float*)d_in[2];  // [4,19,256,256]
    const float* Ave       = (const float*)d_in[3];  // [20,256]
    const float* CoVar     = (const float*)d_in[4];  // [20,256]
    const float* Amount    = (const float*)d_in[5];  // [20]
    const float* ratio     = (const float*)d_in[6];  // [1]
    const int*   tx        = (const int*)d_in[7];    // [4,512,512]
    float* out = (float*)d_out;

    float* wsf      = (float*)d_ws;
    float* ws_sum   = wsf;                         // 5120
    float* ws_sumsq = wsf + KCN * KA;              // 5120
    float* ws_cnt   = wsf + 2 * KCN * KA;          // 20 (pad to 32)
    float* ws_S     = wsf + 2 * KCN * KA + 32;     // 361
    int zero_n = 2 * KCN * KA + 32 + KC * KC;

    zero_ws_kernel<<<32, 256, 0, stream>>>(wsf, zero_n);
    count_kernel<<<256, 256, 0, stream>>>(tx, ws_cnt);
    accum_kernel<<<dim3(KN * KA, 4), 256, 0, stream>>>(features, tx, ws_sum, ws_sumsq);
    build_S_kernel<<<1, 256, 0, stream>>>(fc_weight, Ave, CoVar, Amount, ratio,
                                          ws_sum, ws_sumsq, ws_cnt, ws_S);
    apply_kernel<<<dim3(16, KN * KC), 256, 0, stream>>>(y, tx, ws_S, out);
}